// LossOT_18124761989689
// MI455X (gfx1250) — compile-verified
//
#include <hip/hip_runtime.h>
#include <math.h>

// Problem constants (match reference setup_inputs)
#define BB 2
#define NN 5
#define CC 3
#define HH 48
#define WW 48
#define PP (HH * WW)          // 2304 points per cloud
#define SS 10                 // pairs per batch
#define NPAIR (BB * SS)       // 20 pair losses
#define TILE 16
#define NT (PP / TILE)        // 144 tiles per dimension
#define THREADS 1024
#define NWAVES (THREADS / 32)
#define N_ITERS 20

#define L2E 1.44269504088896340736f   // log2(e)

typedef __attribute__((ext_vector_type(2))) float v2f;
typedef __attribute__((ext_vector_type(8))) float v8f;

// ---------------------------------------------------------------------------
// Kernel 1: diffs[b][n][p] = gt[b][:,p] - burst[b][n][:,p], stored as float4
// (c0,c1,c2,0). The zero pad makes each WMMA fragment a single float2 load.
// ---------------------------------------------------------------------------
__global__ void diffs_kernel(const float* __restrict__ burst,
                             const float* __restrict__ gt,
                             float4* __restrict__ d4, int total) {
  int t = blockIdx.x * blockDim.x + threadIdx.x;
  if (t >= total) return;                 // total = BB*NN*PP
  int p = t % PP;
  int n = (t / PP) % NN;
  int b = t / (PP * NN);
  float c0 = gt[(b * CC + 0) * PP + p] - burst[((b * NN + n) * CC + 0) * PP + p];
  float c1 = gt[(b * CC + 1) * PP + p] - burst[((b * NN + n) * CC + 1) * PP + p];
  float c2 = gt[(b * CC + 2) * PP + p] - burst[((b * NN + n) * CC + 2) * PP + p];
  d4[t] = make_float4(c0, c1, c2, 0.0f);
}

// ---------------------------------------------------------------------------
// One Sinkhorn half-step, entirely in base-2 log domain:
//   logout2[c] = lm2 + 2*L2E*ns[c] - LSE2_r( 4*L2E*dot(Pf[r],Ps[c]) + w2[r] )
// with w2[r] = login2[r] - 2*L2E*nf[r].
// V_WMMA_F32_16X16X4_F32 with A' = (2*L2E)*Pf row frag, B' = 2*Ps col frag,
// C = w2[row] emits the exp2 argument directly (no ln2 fixup per element).
// Each wave owns TWO 16-column strips (shared A and C, two B frags, two
// accumulators); per-lane 8-row online LSE, lane-half combine once per strip.
// ---------------------------------------------------------------------------
__device__ __forceinline__ void sinkhorn_halfstep(
    const float2* __restrict__ Pf, const float2* __restrict__ Ps,
    const float* __restrict__ s_nf, const float* __restrict__ s_ns,
    const float* __restrict__ s_login, float* __restrict__ s_logout,
    float* __restrict__ s_w,
    int tid, int wave, int hi, int l16, float lm2) {
  for (int p = tid; p < PP; p += THREADS)
    s_w[p] = s_login[p] - (2.0f * L2E) * s_nf[p];
  __syncthreads();

  for (int base = 2 * wave; base < NT; base += 2 * NWAVES) {
    const int c0 = base * TILE + l16;
    const int c1 = c0 + TILE;
    // B fragments: lanes 0-15 -> (c0,c1), lanes 16-31 -> (c2,pad). Scale 2.
    float2 b0 = Ps[c0 * 2 + hi];
    float2 b1 = Ps[c1 * 2 + hi];
    v2f bf0; bf0.x = 2.0f * b0.x; bf0.y = 2.0f * b0.y;
    v2f bf1; bf1.x = 2.0f * b1.x; bf1.y = 2.0f * b1.y;

    // Per-lane running LSE over this lane's 8 rows of every tile.
    float m0 = -INFINITY, s0 = 0.0f;
    float m1 = -INFINITY, s1 = 0.0f;

    for (int t = 0; t < NT; ++t) {
      const int rb = t * TILE;
      float2 af = Pf[(rb + l16) * 2 + hi];
      v2f afrag; afrag.x = (2.0f * L2E) * af.x; afrag.y = (2.0f * L2E) * af.y;

      // C = w2[row]: shared by both strips (same rows).
      const float4* wq = (const float4*)(s_w + rb + hi * 8);
      float4 w0 = wq[0], w1 = wq[1];
      v8f c;
      c[0] = w0.x; c[1] = w0.y; c[2] = w0.z; c[3] = w0.w;
      c[4] = w1.x; c[5] = w1.y; c[6] = w1.z; c[7] = w1.w;

      v8f acc0 = __builtin_amdgcn_wmma_f32_16x16x4_f32(
          false, afrag, false, bf0, (short)0, c, false, false);
      v8f acc1 = __builtin_amdgcn_wmma_f32_16x16x4_f32(
          false, afrag, false, bf1, (short)0, c, false, false);

      // Strip 0: in-lane 8-value LSE tile update (base-2).
      {
        float tmx = acc0[0];
        #pragma unroll
        for (int v = 1; v < 8; ++v) tmx = fmaxf(tmx, acc0[v]);
        float ts = 0.0f;
        #pragma unroll
        for (int v = 0; v < 8; ++v) ts += __builtin_amdgcn_exp2f(acc0[v] - tmx);
        float nm = fmaxf(m0, tmx);
        s0 = s0 * __builtin_amdgcn_exp2f(m0 - nm)
           + ts * __builtin_amdgcn_exp2f(tmx - nm);
        m0 = nm;
      }
      // Strip 1.
      {
        float tmx = acc1[0];
        #pragma unroll
        for (int v = 1; v < 8; ++v) tmx = fmaxf(tmx, acc1[v]);
        float ts = 0.0f;
        #pragma unroll
        for (int v = 0; v < 8; ++v) ts += __builtin_amdgcn_exp2f(acc1[v] - tmx);
        float nm = fmaxf(m1, tmx);
        s1 = s1 * __builtin_amdgcn_exp2f(m1 - nm)
           + ts * __builtin_amdgcn_exp2f(tmx - nm);
        m1 = nm;
      }
    }

    // Combine the two lane halves (rows 0-7 vs 8-15) once per strip.
    {
      float om = __shfl_xor(m0, 16, 32);
      float os = __shfl_xor(s0, 16, 32);
      float M  = fmaxf(m0, om);
      float S  = s0 * __builtin_amdgcn_exp2f(m0 - M)
               + os * __builtin_amdgcn_exp2f(om - M);
      float lse = M + __builtin_amdgcn_logf(S);
      s_logout[c0] = lm2 + (2.0f * L2E) * s_ns[c0] - lse;  // dup write, same val
    }
    {
      float om = __shfl_xor(m1, 16, 32);
      float os = __shfl_xor(s1, 16, 32);
      float M  = fmaxf(m1, om);
      float S  = s1 * __builtin_amdgcn_exp2f(m1 - M)
               + os * __builtin_amdgcn_exp2f(om - M);
      float lse = M + __builtin_amdgcn_logf(S);
      s_logout[c1] = lm2 + (2.0f * L2E) * s_ns[c1] - lse;
    }
  }
  __syncthreads();
}

// ---------------------------------------------------------------------------
// Kernel 2: one workgroup per (b, i, j) pair. 32 waves, 1024 threads.
// ---------------------------------------------------------------------------
__launch_bounds__(THREADS, 1)
__global__ void sinkhorn_pair_kernel(const float4* __restrict__ diffs,
                                     const int* __restrict__ pairs,
                                     float* __restrict__ losses) {
  __shared__ float s_ni[PP];
  __shared__ float s_nj[PP];
  __shared__ float s_logu[PP];    // base-2 scaled potentials
  __shared__ float s_logv[PP];
  __shared__ float s_w[PP];
  __shared__ float s_red[NWAVES];

  const int tid  = threadIdx.x;
  const int lane = tid & 31;
  const int wave = tid >> 5;
  const int hi   = lane >> 4;     // lane half: selects (c0,c1) vs (c2,pad)
  const int l16  = lane & 15;

  const int pairId = blockIdx.x;            // = b*SS + s
  const int b  = pairId / SS;
  const int pi = pairs[pairId * 2 + 0];
  const int pj = pairs[pairId * 2 + 1];
  const float2* Pi = (const float2*)(diffs + (b * NN + pi) * PP);
  const float2* Pj = (const float2*)(diffs + (b * NN + pj) * PP);
  const float lm2 = -__builtin_amdgcn_logf((float)PP);  // log2(1/n)

  // Point norms (natural units) + marginal init (base-2).
  for (int p = tid; p < PP; p += THREADS) {
    float2 il = Pi[p * 2 + 0], ih = Pi[p * 2 + 1];
    float2 jl = Pj[p * 2 + 0], jh = Pj[p * 2 + 1];
    s_ni[p] = il.x * il.x + il.y * il.y + ih.x * ih.x;
    s_nj[p] = jl.x * jl.x + jl.y * jl.y + jh.x * jh.x;
    s_logu[p] = lm2;
    s_logv[p] = lm2;
  }
  __syncthreads();

  for (int it = 0; it < N_ITERS; ++it) {
    // logv update (reduce over i), then logu update (reduce over j).
    sinkhorn_halfstep(Pi, Pj, s_ni, s_nj, s_logu, s_logv, s_w,
                      tid, wave, hi, l16, lm2);
    sinkhorn_halfstep(Pj, Pi, s_nj, s_ni, s_logv, s_logu, s_w,
                      tid, wave, hi, l16, lm2);
  }

  // Final pass: loss = sum exp2(logu2 - 2*L2E*ni + 4*L2E*dot + logv2
  //                            - 2*L2E*nj) * (ni + nj - 2*dot)
  for (int p = tid; p < PP; p += THREADS)
    s_w[p] = s_logu[p] - (2.0f * L2E) * s_ni[p];
  __syncthreads();

  float partial = 0.0f;
  for (int base = 2 * wave; base < NT; base += 2 * NWAVES) {
    const int c0 = base * TILE + l16;
    const int c1 = c0 + TILE;
    float2 b0 = Pj[c0 * 2 + hi];
    float2 b1 = Pj[c1 * 2 + hi];
    v2f bf0; bf0.x = b0.x; bf0.y = b0.y;    // unscaled: acc = raw dot
    v2f bf1; bf1.x = b1.x; bf1.y = b1.y;
    const float cv0 = s_logv[c0] - (2.0f * L2E) * s_nj[c0];
    const float cv1 = s_logv[c1] - (2.0f * L2E) * s_nj[c1];
    const float nj0 = s_nj[c0];
    const float nj1 = s_nj[c1];

    for (int t = 0; t < NT; ++t) {
      const int rb = t * TILE;
      float2 af = Pi[(rb + l16) * 2 + hi];
      v2f afrag; afrag.x = af.x; afrag.y = af.y;
      v8f z = {0.f, 0.f, 0.f, 0.f, 0.f, 0.f, 0.f, 0.f};
      v8f acc0 = __builtin_amdgcn_wmma_f32_16x16x4_f32(
          false, afrag, false, bf0, (short)0, z, false, false);
      v8f acc1 = __builtin_amdgcn_wmma_f32_16x16x4_f32(
          false, afrag, false, bf1, (short)0, z, false, false);

      const float4* wq = (const float4*)(s_w + rb + hi * 8);
      const float4* nq = (const float4*)(s_ni + rb + hi * 8);
      float4 w0 = wq[0], w1 = wq[1], n0 = nq[0], n1 = nq[1];
      float wv[8] = {w0.x, w0.y, w0.z, w0.w, w1.x, w1.y, w1.z, w1.w};
      float nv[8] = {n0.x, n0.y, n0.z, n0.w, n1.x, n1.y, n1.z, n1.w};
      #pragma unroll
      for (int v = 0; v < 8; ++v) {
        float d   = acc0[v];
        float lg2 = wv[v] + (4.0f * L2E) * d + cv0;   // log2(gamma)
        float Mij = nv[v] + nj0 - 2.0f * d;
        partial += __builtin_amdgcn_exp2f(lg2) * Mij;
      }
      #pragma unroll
      for (int v = 0; v < 8; ++v) {
        float d   = acc1[v];
        float lg2 = wv[v] + (4.0f * L2E) * d + cv1;
        float Mij = nv[v] + nj1 - 2.0f * d;
        partial += __builtin_amdgcn_exp2f(lg2) * Mij;
      }
    }
  }

  // Block reduction.
  #pragma unroll
  for (int off = 16; off >= 1; off >>= 1)
    partial += __shfl_xor(partial, off, 32);
  if (lane == 0) s_red[wave] = partial;
  __syncthreads();
  if (wave == 0) {
    float v = s_red[lane];
    #pragma unroll
    for (int off = 16; off >= 1; off >>= 1)
      v += __shfl_xor(v, off, 32);
    if (lane == 0) losses[pairId] = v;
  }
}

// ---------------------------------------------------------------------------
// Kernel 3: mean of the 20 pair losses -> d_out[0] (overwrite, no init needed)
// ---------------------------------------------------------------------------
__global__ void finalize_kernel(const float* __restrict__ losses,
                                float* __restrict__ out) {
  float v = (threadIdx.x < NPAIR) ? losses[threadIdx.x] : 0.0f;
  #pragma unroll
  for (int off = 16; off >= 1; off >>= 1)
    v += __shfl_xor(v, off, 32);
  if (threadIdx.x == 0) out[0] = v * (1.0f / (float)NPAIR);
}

extern "C" void kernel_launch(void* const* d_in, const int* in_sizes, int n_in,
                              void* d_out, int out_size, void* d_ws, size_t ws_size,
                              hipStream_t stream) {
  const float* burst = (const float*)d_in[0];   // [B,N,C,H,W] f32
  const float* gt    = (const float*)d_in[1];   // [B,C,H,W]  f32
  const int*   pairs = (const int*)d_in[2];     // [B,S,2]    int32
  float* out = (float*)d_out;

  // Workspace: padded diffs (float4 per point) then per-pair losses.
  float4* diffs  = (float4*)d_ws;
  float*  losses = (float*)((char*)d_ws + (size_t)BB * NN * PP * sizeof(float4));

  const int total = BB * NN * PP;
  diffs_kernel<<<(total + 255) / 256, 256, 0, stream>>>(burst, gt, diffs, total);
  sinkhorn_pair_kernel<<<NPAIR, THREADS, 0, stream>>>(diffs, pairs, losses);
  finalize_kernel<<<1, 32, 0, stream>>>(losses, out);
}